// CMSMemoryModule_63050119905754
// MI455X (gfx1250) — compile-verified
//
#include <hip/hip_runtime.h>
#include <hip/hip_bf16.h>

// ---------------- problem constants ----------------
#define BB   16
#define CC   256
#define HW   9216           // 96*96
#define MD   64
#define NH   4
#define HD   16
#define GG   8
#define CPG  32             // C / G
#define NT   32             // n-tile (columns) per block
#define NTILES 288          // HW / NT
#define EPS  1e-5f

// ---------------- workspace layout (float offsets) ----------------
#define WS_CHSUM 0                         // B*C          = 4096
#define WS_CHSQ  4096                      // B*C          = 4096
#define WS_MU    8192                      // B*G          = 128
#define WS_RSIG  8320                      // B*G          = 128
#define WS_GATE  8448                      // B            = 16
#define WS_SC    8464                      // B*C          = 4096 (per-channel scale)
#define WS_SH    12560                     // B*C          = 4096 (per-channel shift)
#define WS_M     16656                     // B*NH*HD*HD   = 16384
#define WS_Z     33040                     // B*NH*HD      = 1024
#define WS_MP    34064                     // (B*NTILES)*1088 = 5013504
#define WS_PQ    5047568ull                // B*MD*HW      = 9437184
// total ~14.49M floats (~58 MB)

typedef float v2f __attribute__((ext_vector_type(2)));
typedef float v8f __attribute__((ext_vector_type(8)));

__device__ __forceinline__ v8f wmma4(v2f a, v2f b, v8f c) {
  // V_WMMA_F32_16X16X4_F32 : D(16x16 f32) = A(16x4 f32) x B(4x16 f32) + C
  return __builtin_amdgcn_wmma_f32_16x16x4_f32(false, a, false, b, (short)0, c, false, false);
}

// phi(x) = elu(x)+1, branchless: exp computed unconditionally -> v_cndmask, no saveexec
__device__ __forceinline__ float phi(float x) {
  float e = __expf(x);
  return x > 0.0f ? x + 1.0f : e;
}

// ================= pass 1: per-(b,c) sum / sumsq over HW =================
__global__ __launch_bounds__(256) void k_stats(const float* __restrict__ x,
                                               float* __restrict__ ws) {
  int bc = blockIdx.x;                     // 0 .. B*C-1
  const float* p = x + (size_t)bc * HW;
  float s = 0.f, s2 = 0.f;
  for (int i = threadIdx.x; i < HW; i += 256) {
    float v = p[i];
    s += v; s2 += v * v;
  }
  __shared__ float rs[256], rq[256];
  int t = threadIdx.x;
  rs[t] = s; rq[t] = s2;
  __syncthreads();
  for (int off = 128; off > 0; off >>= 1) {
    if (t < off) { rs[t] += rs[t + off]; rq[t] += rq[t + off]; }
    __syncthreads();
  }
  if (t == 0) { ws[WS_CHSUM + bc] = rs[0]; ws[WS_CHSQ + bc] = rq[0]; }
}

// ==== pass 1b: group stats + per-channel affine + pooled + gate (per batch) ====
__global__ __launch_bounds__(256) void k_finalize(const float* __restrict__ nw,
                                                  const float* __restrict__ nb,
                                                  const float* __restrict__ Wg1,
                                                  const float* __restrict__ bg1,
                                                  const float* __restrict__ Wg2,
                                                  const float* __restrict__ bg2,
                                                  float* __restrict__ ws,
                                                  float* __restrict__ d_out) {
  int b = blockIdx.x, t = threadIdx.x;
  __shared__ float pooled[CC];
  __shared__ float gmu[GG], grs[GG];
  __shared__ float t1[64];
  if (t < GG) {
    float s = 0.f, s2 = 0.f;
    for (int c = t * CPG; c < (t + 1) * CPG; ++c) {
      s  += ws[WS_CHSUM + b * CC + c];
      s2 += ws[WS_CHSQ  + b * CC + c];
    }
    float inv = 1.0f / ((float)CPG * (float)HW);
    float mu  = s * inv;
    float var = s2 * inv - mu * mu;
    float rsg = rsqrtf(var + EPS);
    gmu[t] = mu; grs[t] = rsg;
    ws[WS_MU   + b * GG + t] = mu;
    ws[WS_RSIG + b * GG + t] = rsg;
  }
  __syncthreads();
  {  // per-channel affine for xn, and pooled[c] = mean_HW(xn[c])
    int c = t, g = c >> 5;
    float sc = grs[g] * nw[c];
    float sh = nb[c] - gmu[g] * sc;
    ws[WS_SC + b * CC + c] = sc;
    ws[WS_SH + b * CC + c] = sh;
    float m = ws[WS_CHSUM + b * CC + c] * (1.0f / (float)HW);
    pooled[c] = m * sc + sh;
  }
  __syncthreads();
  if (t < 64) {
    float acc = bg1[t];
    for (int c = 0; c < CC; ++c) acc += Wg1[t * CC + c] * pooled[c];
    float ge = 0.5f * acc * (1.0f + erff(acc * 0.70710678118654752f));  // exact GELU
    t1[t] = ge * Wg2[t];
  }
  __syncthreads();
  if (t == 0) {
    float g = bg2[0];
    for (int j = 0; j < 64; ++j) g += t1[j];
    float gate = 1.0f / (1.0f + __expf(-g));
    ws[WS_GATE + b] = gate;
    d_out[(size_t)BB * CC * HW + b] = gate;   // second tuple output
  }
}

// ==== pass 2: xn -> fused kvq GEMM (WMMA f32) -> phi -> M/Z partials + phi_q ====
__global__ __launch_bounds__(256) void k_proj(const float* __restrict__ x,
                                              const float* __restrict__ Wk,
                                              const float* __restrict__ Wv,
                                              const float* __restrict__ Wq,
                                              float* __restrict__ ws) {
  __shared__ float xn_s[CC * 33];      // 256 x 32 (+pad)
  __shared__ float kvq_s[192 * 33];    // rows: [k 0..63][v 64..127][q 128..191]
  int blk = blockIdx.x;
  int batch = blk / NTILES, tile = blk % NTILES;
  int tid = threadIdx.x, lane = tid & 31, wave = tid >> 5;
  int m = lane & 15, hi = lane >> 4, koff = 2 * hi;

  const float* scv = ws + WS_SC + batch * CC;
  const float* shv = ws + WS_SH + batch * CC;

  // load x tile, normalize on the fly (single fma per element)
  const float* xb = x + (size_t)batch * CC * HW + (size_t)tile * NT;
  for (int i = tid; i < CC * NT; i += 256) {
    int c = i >> 5, col = i & 31;
    xn_s[c * 33 + col] = fmaf(xb[(size_t)c * HW + col], scv[c], shv[c]);
  }
  __syncthreads();

  // kvq(192x32) = [Wk;Wv;Wq](192x256) @ xn(256x32) ; 24 wmma tiles over 8 waves
  for (int t = wave; t < 24; t += 8) {
    int rt = t >> 1, ct = t & 1;
    const float* Wsel = (rt < 4) ? Wk : ((rt < 8) ? Wv : Wq);
    const float* arow = Wsel + ((rt & 3) * 16 + m) * CC + koff;
    v8f acc = {};
    for (int k0 = 0; k0 < CC; k0 += 4) {
      v2f a;  a.x = arow[k0];  a.y = arow[k0 + 1];
      v2f bf; bf.x = xn_s[(k0 + koff) * 33 + ct * 16 + m];
              bf.y = xn_s[(k0 + koff + 1) * 33 + ct * 16 + m];
      acc = wmma4(a, bf, acc);
    }
    bool isphi = (rt < 4) || (rt >= 8);          // phi on k and q rows
    for (int r = 0; r < 8; ++r) {
      float v = acc[r];
      if (isphi) v = phi(v);
      kvq_s[(rt * 16 + r + 8 * hi) * 33 + ct * 16 + m] = v;
    }
  }
  __syncthreads();

  float* Mp = ws + WS_MP + (size_t)blk * 1088;   // per-block partial [4*256 M | 64 Z]
  if (wave < 4) {
    // M_local[d][e] = sum_n phi_k[d][n] * v[e][n]   (K = n, 8 wmma steps)
    int h = wave;
    const float* pk = kvq_s + (h * 16 + m) * 33;        // A row d = m
    const float* pv = kvq_s + (64 + h * 16 + m) * 33;   // B "N" lane e = m
    v8f macc = {};
    for (int n0 = 0; n0 < NT; n0 += 4) {
      v2f a;  a.x = pk[n0 + koff];  a.y = pk[n0 + koff + 1];
      v2f bf; bf.x = pv[n0 + koff]; bf.y = pv[n0 + koff + 1];
      macc = wmma4(a, bf, macc);
    }
    for (int r = 0; r < 8; ++r)
      Mp[h * 256 + (r + 8 * hi) * 16 + m] = macc[r];
  } else {
    int t2 = tid - 128;                          // 0..127
    if (t2 < 64) {                               // Z partial: row sums of phi_k
      const float* pk = kvq_s + t2 * 33;
      float s = 0.f;
      for (int c2 = 0; c2 < NT; ++c2) s += pk[c2];
      Mp[1024 + t2] = s;
    }
    // spill phi_q tile (rows 128..191) to global ws
    float* pq = ws + WS_PQ + (size_t)batch * MD * HW + (size_t)tile * NT;
    for (int i = t2; i < MD * NT; i += 128) {
      int r2 = i >> 5, c2 = i & 31;
      pq[(size_t)r2 * HW + c2] = kvq_s[(128 + r2) * 33 + c2];
    }
  }
}

// ===== deterministic reduction of M/Z partials over tiles, + M0/Z0 bias =====
__global__ __launch_bounds__(256) void k_reduceMZ(const float* __restrict__ M0,
                                                  const float* __restrict__ Z0,
                                                  float* __restrict__ ws) {
  int b = blockIdx.x >> 2, h = blockIdx.x & 3;   // 64 blocks
  int t = threadIdx.x;
  const float* base = ws + WS_MP;
  float s = 0.f;
  for (int i = 0; i < NTILES; ++i)
    s += base[(size_t)(b * NTILES + i) * 1088 + h * 256 + t];
  ws[WS_M + (b * NH + h) * 256 + t] = M0[h * 256 + t] + s;
  if (t < HD) {
    float z = 0.f;
    for (int i = 0; i < NTILES; ++i)
      z += base[(size_t)(b * NTILES + i) * 1088 + 1024 + h * HD + t];
    ws[WS_Z + (b * NH + h) * HD + t] = Z0[h * HD + t] + z;
  }
}

// ==== pass 3: y = (M^T phi_q)/clip(Z.phi_q) -> Wout GEMM -> gated residual ====
__global__ __launch_bounds__(256) void k_out(const float* __restrict__ x,
                                             const float* __restrict__ Wout,
                                             const float* __restrict__ bout,
                                             const float* __restrict__ ws,
                                             float* __restrict__ d_out) {
  __shared__ float pq_s[MD * 33];
  __shared__ float y_s[MD * 33];
  __shared__ float Msh[NH * 256];
  __shared__ float Zsh[MD];
  int blk = blockIdx.x;
  int batch = blk / NTILES, tile = blk % NTILES;
  int tid = threadIdx.x, lane = tid & 31, wave = tid >> 5;
  int m = lane & 15, hi = lane >> 4, koff = 2 * hi;

  for (int i = tid; i < NH * 256; i += 256) Msh[i] = ws[WS_M + batch * NH * 256 + i];
  if (tid < MD) Zsh[tid] = ws[WS_Z + batch * MD + tid];
  const float* pqg = ws + WS_PQ + (size_t)batch * MD * HW + (size_t)tile * NT;
  for (int i = tid; i < MD * NT; i += 256) {
    int r = i >> 5, c = i & 31;
    pq_s[r * 33 + c] = pqg[(size_t)r * HW + c];
  }
  float gate = ws[WS_GATE + batch];
  __syncthreads();

  // y_num[e][n] = sum_d M[d][e] phi_q[d][n] ; one (head, col-tile) per wave
  {
    int h = wave >> 1, ct = wave & 1;
    const float* Mh = Msh + h * 256;
    v8f acc = {};
    for (int k0 = 0; k0 < HD; k0 += 4) {
      v2f a;  a.x = Mh[(k0 + koff) * 16 + m];          // A[e=m][d] = M[d][e]
              a.y = Mh[(k0 + koff + 1) * 16 + m];
      v2f bf; bf.x = pq_s[(h * 16 + k0 + koff) * 33 + ct * 16 + m];
              bf.y = pq_s[(h * 16 + k0 + koff + 1) * 33 + ct * 16 + m];
      acc = wmma4(a, bf, acc);
    }
    float den = 0.f;                                   // den for column n = ct*16+m
    for (int d = 0; d < HD; ++d)
      den += Zsh[h * 16 + d] * pq_s[(h * 16 + d) * 33 + ct * 16 + m];
    float invd = 1.0f / fmaxf(den, 1e-4f);
    for (int r = 0; r < 8; ++r)
      y_s[(h * 16 + r + 8 * hi) * 33 + ct * 16 + m] = acc[r] * invd;
  }
  __syncthreads();

  // out(256x32) = Wout(256x64) @ y(64x32) ; fused bias + gate + residual
  const float* xb = x   + (size_t)batch * CC * HW + (size_t)tile * NT;
  float* ob       = d_out + (size_t)batch * CC * HW + (size_t)tile * NT;
  for (int t = wave; t < 32; t += 8) {
    int rt = t >> 1, ct = t & 1;
    const float* arow = Wout + (rt * 16 + m) * MD + koff;
    v8f acc = {};
    for (int k0 = 0; k0 < MD; k0 += 4) {
      v2f a;  a.x = arow[k0];  a.y = arow[k0 + 1];
      v2f bf; bf.x = y_s[(k0 + koff) * 33 + ct * 16 + m];
              bf.y = y_s[(k0 + koff + 1) * 33 + ct * 16 + m];
      acc = wmma4(a, bf, acc);
    }
    for (int r = 0; r < 8; ++r) {
      int crow = rt * 16 + r + 8 * hi;
      size_t idx = (size_t)crow * HW + ct * 16 + m;
      ob[idx] = xb[idx] + gate * (acc[r] + bout[crow]);
    }
  }
}

// ---------------------------------------------------------------------------
extern "C" void kernel_launch(void* const* d_in, const int* in_sizes, int n_in,
                              void* d_out, int out_size, void* d_ws, size_t ws_size,
                              hipStream_t stream) {
  const float* x    = (const float*)d_in[0];
  const float* nw   = (const float*)d_in[1];
  const float* nb   = (const float*)d_in[2];
  const float* Wk   = (const float*)d_in[3];
  const float* Wv   = (const float*)d_in[4];
  const float* Wq   = (const float*)d_in[5];
  const float* Wout = (const float*)d_in[6];
  const float* bout = (const float*)d_in[7];
  const float* M0   = (const float*)d_in[8];
  const float* Z0   = (const float*)d_in[9];
  const float* Wg1  = (const float*)d_in[10];
  const float* bg1  = (const float*)d_in[11];
  const float* Wg2  = (const float*)d_in[12];
  const float* bg2  = (const float*)d_in[13];
  float* out = (float*)d_out;
  float* ws  = (float*)d_ws;

  k_stats   <<<BB * CC,       256, 0, stream>>>(x, ws);
  k_finalize<<<BB,            256, 0, stream>>>(nw, nb, Wg1, bg1, Wg2, bg2, ws, out);
  k_proj    <<<BB * NTILES,   256, 0, stream>>>(x, Wk, Wv, Wq, ws);
  k_reduceMZ<<<BB * NH,       256, 0, stream>>>(M0, Z0, ws);
  k_out     <<<BB * NTILES,   256, 0, stream>>>(x, Wout, bout, ws, out);
}